// PRMS_10849087389888
// MI455X (gfx1250) — compile-verified
//
#include <hip/hip_runtime.h>

#define T_STEPS 730
#define N_GRID  8000
#define NEARZERO 1e-5f

__device__ __forceinline__ float rcp_f(float x)  { return __builtin_amdgcn_rcpf(x);  }  // v_rcp_f32
__device__ __forceinline__ float ex2_f(float x)  { return __builtin_amdgcn_exp2f(x); }  // v_exp_f32
__device__ __forceinline__ float lg2_f(float x)  { return __builtin_amdgcn_logf(x);  }  // v_log_f32
__device__ __forceinline__ float clamp01(float x){ return __builtin_amdgcn_fmed3f(x, 0.0f, 1.0f); } // v_med3_f32

__global__ __launch_bounds__(64)
void prms_scan_kernel(const float* __restrict__ x,
                      const float* __restrict__ params,
                      float* __restrict__ out)
{
    const int c = blockIdx.x * blockDim.x + threadIdx.x;
    if (c >= N_GRID) return;

    // ---- per-cell parameter derivation (prologue, runs once) ----
    const float lo[18] = {-3.f, 0.f, 0.f, 0.f, 0.f, 0.f, 0.f, 0.f, 0.005f,
                           1.f, 0.f, 1.f, 0.f, 1.f, 0.f, 0.f, 0.f, 0.f};
    const float hi[18] = { 5.f, 20.f, 1.f, 1.f, 5.f, 50.f, 1.f, 1.f, 0.995f,
                           2000.f, 20.f, 300.f, 1.f, 5.f, 1.f, 1.f, 1.f, 1.f};

    const float* __restrict__ pr =
        params + (size_t)(T_STEPS - 1) * N_GRID * 18 + (size_t)c * 18;

    float p[18];
#pragma unroll
    for (int i = 0; i < 18; ++i) {
        // sigmoid(z) = 1/(1+exp(-z)) = rcp(1 + exp2(-z*log2e))
        float s = rcp_f(1.0f + ex2_f(-pr[i] * 1.4426950408889634f));
        p[i] = lo[i] + s * (hi[i] - lo[i]);
    }
    const float tt = p[0],  ddf = p[1],  alpha = p[2],  beta = p[3];
    const float stor = p[4], retip = p[5], fscn = p[6], scx = p[7];
    const float flz = p[8],  stot = p[9], cgw = p[10],  resmax = p[11];
    const float k1 = p[12],  k2 = p[13], k3 = p[14],   k4 = p[15];
    const float k5 = p[16],  k6 = p[17];

    const float scn        = fscn * scx;
    const float remx       = (1.0f - flz) * stot;
    const float smax       = flz * stot;
    const float inv_remx   = rcp_f(remx);      // divisions hoisted out of the loop
    const float inv_smax   = rcp_f(smax);
    const float inv_resmax = rcp_f(resmax);
    const float one_m_beta = 1.0f - beta;
    const float one_m_alpha= 1.0f - alpha;
    const float scx_m_scn  = scx - scn;

    // ---- state ----
    float snow = 0.001f, xin = 0.001f, rstor = 0.001f, rechr = 0.001f;
    float smav = 0.001f, res = 0.001f, gw = 0.001f;

    const float* __restrict__ xp = x + (size_t)c * 3;
    const size_t xstride = (size_t)N_GRID * 3;
    float* __restrict__ qp = out + c;

    // register double-buffer of forcing: load for t=0 up front (single-use
    // stream -> non-temporal so it doesn't displace anything in L2)
    float P  = __builtin_nontemporal_load(xp + 0);
    float Tm = __builtin_nontemporal_load(xp + 1);
    float Ep = __builtin_nontemporal_load(xp + 2);

    for (int t = 0; t < T_STEPS; ++t) {
        // long-range prefetch of forcing stream (gfx1250 global_prefetch_b8)
        const int tpf = (t + 24 < T_STEPS) ? (t + 24) : (T_STEPS - 1);
        __builtin_prefetch(xp + (size_t)tpf * xstride, 0, 0);

        // issue next timestep's loads early; they are independent of the state
        // chain, so VMEM latency overlaps the dependent VALU work below.
        const int tn = (t + 1 < T_STEPS) ? (t + 1) : (T_STEPS - 1);
        const float* __restrict__ xn = xp + (size_t)tn * xstride;
        const float Pn = __builtin_nontemporal_load(xn + 0);
        const float Tn = __builtin_nontemporal_load(xn + 1);
        const float En = __builtin_nontemporal_load(xn + 2);

        // ---- snow pack ----
        const float flux_ps = (Tm <= tt) ? P : 0.0f;
        const float flux_pr = P - flux_ps;
        snow += flux_ps;
        float flux_m = fmaxf(ddf * (Tm - tt), 0.0f);
        flux_m = fminf(flux_m, snow);
        snow = fmaxf(snow - flux_m, NEARZERO);

        // ---- interception ----
        const float flux_pim = flux_pr * one_m_beta;
        const float flux_psm = flux_pr * beta;
        const float flux_pby = flux_psm * one_m_alpha;
        const float flux_pin = flux_psm * alpha;
        xin += flux_pin;
        const float flux_ptf = fmaxf(xin - stor, 0.0f);
        xin = fmaxf(xin - flux_ptf, NEARZERO);
        const float flux_ein = fminf(Ep * beta, xin);
        xin = fmaxf(xin - flux_ein, NEARZERO);

        // ---- impervious store ----
        const float flux_mim = flux_m * one_m_beta;
        const float flux_msm = flux_m * beta;
        rstor += flux_mim + flux_pim;
        const float flux_sas = fmaxf(rstor - retip, 0.0f);
        rstor = fmaxf(rstor - flux_sas, NEARZERO);
        const float flux_eim = fminf(one_m_beta * Ep, rstor);
        rstor = fmaxf(rstor - flux_eim, NEARZERO);

        // ---- surface runoff / recharge zone ----
        const float sro_lin = clamp01(scn + scx_m_scn * (rechr * inv_remx)); // v_med3_f32
        const float inflow   = flux_msm + flux_ptf + flux_pby;
        const float flux_sro = sro_lin * inflow;
        const float flux_inf = inflow - flux_sro;
        rechr += flux_inf;
        const float flux_pc = fmaxf(rechr - remx, 0.0f);
        rechr -= flux_pc;
        const float ep_rem = Ep - flux_ein - flux_eim;
        const float evap_max_a = fmaxf(rechr * inv_remx * ep_rem, 0.0f);
        const float flux_ea = fminf(evap_max_a, rechr);
        rechr = fmaxf(rechr - flux_ea, NEARZERO);

        // ---- lower soil zone ----
        smav += flux_pc;
        const float flux_excs = fmaxf(smav - smax, 0.0f);
        smav -= flux_excs;
        float transp = (rechr < ep_rem)
                         ? fmaxf(smav * inv_smax * (ep_rem - flux_ea), 0.0f)
                         : 0.0f;
        transp = fminf(transp, smav);
        smav = fmaxf(smav - transp, NEARZERO);

        // ---- subsurface reservoir ----
        const float flux_sep  = fminf(cgw, flux_excs);
        const float flux_qres = fmaxf(flux_excs - flux_sep, 0.0f);
        res += flux_qres;
        // k1 * (res/resmax)^k2  ==  k1 * exp2(k2 * log2(res*inv_resmax)), res >= NEARZERO > 0
        float flux_gad = k1 * ex2_f(k2 * lg2_f(res * inv_resmax));
        flux_gad = fminf(flux_gad, res);
        res = fmaxf(res - flux_gad, NEARZERO);
        float flux_ras = k3 * res + k4 * res * res;
        flux_ras = fminf(flux_ras, res);
        res = fmaxf(res - flux_ras, NEARZERO);

        // ---- groundwater ----
        gw += flux_gad + flux_sep;
        const float flux_bas = k5 * gw;
        gw = fmaxf(gw - flux_bas, NEARZERO);
        const float flux_snk = k6 * gw;
        gw = fmaxf(gw - flux_snk, NEARZERO);

        // ---- streamflow (single-use output stream -> non-temporal store) ----
        const float q = flux_sas + flux_sro + flux_bas + flux_ras;
        __builtin_nontemporal_store(q, qp + (size_t)t * N_GRID);

        P = Pn; Tm = Tn; Ep = En;
    }
}

extern "C" void kernel_launch(void* const* d_in, const int* in_sizes, int n_in,
                              void* d_out, int out_size, void* d_ws, size_t ws_size,
                              hipStream_t stream)
{
    const float* x      = (const float*)d_in[0];   // (730, 8000, 3) float32
    const float* params = (const float*)d_in[1];   // (730, 8000, 18) float32 (only row 729 used)
    float* out          = (float*)d_out;           // (730, 8000) float32

    (void)in_sizes; (void)n_in; (void)out_size; (void)d_ws; (void)ws_size;

    // Small blocks -> more workgroups -> waves spread over many WGPs.
    // The scan is latency-bound per wave, so each wave wants its own SIMD.
    const int block = 64;                          // 2 wave32 waves per workgroup
    const int grid  = (N_GRID + block - 1) / block; // 125 workgroups
    prms_scan_kernel<<<grid, block, 0, stream>>>(x, params, out);
}